// Net_61667140436406
// MI455X (gfx1250) — compile-verified
//
#include <hip/hip_runtime.h>
#include <hip/hip_bf16.h>
#include <math.h>

#define NFACE 8

typedef __attribute__((ext_vector_type(2))) float v2f;
typedef __attribute__((ext_vector_type(8))) float v8f;

// One K-step (4 K-elements) of the WMMA GEMV/GEMM pipeline. Branch-free.
template <bool PREFETCH>
__device__ __forceinline__ void wmma_step(const float*& Xp, const float*& Wp,
                                          size_t Ns, float am, float bm, v8f& acc)
{
    v2f a;
    a.x = Xp[0] * am;
    a.y = Xp[1] * am;
    v2f b;
    b.x = Wp[0]  * bm;
    b.y = Wp[Ns] * bm;
    if (PREFETCH)
        __builtin_prefetch(Wp + 32 * Ns, 0, 1);   // global_prefetch_b8
    acc = __builtin_amdgcn_wmma_f32_16x16x4_f32(
              false, a, false, b, (short)0, acc, false, false);
    Xp += 4;
    Wp += 4 * Ns;
}

// ---------------------------------------------------------------------------
// WMMA-based skinny GEMM / GEMV:  Y[M x N] = act(X[M x K] @ W[K x N] + bias)
// One wave owns a 16-column tile of Y; K consumed 4 per V_WMMA_F32_16X16X4_F32.
// fp32 weights are streamed directly (runtime bf16 conversion would double
// HBM traffic; the whole net is a 1.05 GB weight stream -> ~45us at 23.3TB/s).
//
// Inner loop is branch-free: lane validity via clamped addresses + 0/1 masks,
// pointer strength reduction, unroll x4, and loop fission so the prefetch
// needs no per-step guard branch.
//
// A (16x4 f32) lane layout (ISA 7.12.2): lane m (m<16) holds K=k,k+1 in the
// two A VGPRs; lane 16+m holds K=k+2,k+3.  B (4x16) mirrors that K striping.
// D: VGPR j = row M=j on lanes 0-15, row M=8+j on lanes 16-31.
// ---------------------------------------------------------------------------
__global__ void gemm_wmma_f32(const float* __restrict__ X,
                              const float* __restrict__ W,
                              const float* __restrict__ bias,
                              float* __restrict__ Y,
                              int M, int K, int N, int do_relu)
{
    const int lane = threadIdx.x & 31;
    const int wave = threadIdx.x >> 5;
    const int tile = blockIdx.x * (blockDim.x >> 5) + wave;
    const int col0 = tile * 16;
    if (col0 >= N) return;                 // wave-uniform exit: EXEC stays full

    const int mrow = lane & 15;
    const int kofs = (lane >> 4) << 1;     // 0 for lanes 0-15, 2 for lanes 16-31
    const int n    = col0 + (lane & 15);

    // lane-validity as multiplier masks + clamped (always-legal) addresses
    const int   mc = (mrow < M) ? mrow : 0;
    const float am = (mrow < M) ? 1.f : 0.f;
    const int   nc = (n < N) ? n : 0;
    const float bm = (n < N) ? 1.f : 0.f;

    const float* Xp = X + (size_t)mc * K + kofs;   // advances by 4
    const float* Wp = W + (size_t)kofs * N + nc;   // advances by 4*N
    const size_t Ns = (size_t)N;

    v8f acc = {0.f, 0.f, 0.f, 0.f, 0.f, 0.f, 0.f, 0.f};

    const int K4  = K & ~3;
    const int Kpf = (K4 > 36) ? (K4 - 36) : 0;  // steps allowed to prefetch +32 rows

    int k = 0;
    #pragma unroll 4
    for (; k < Kpf; k += 4)                     // main: unconditional prefetch
        wmma_step<true>(Xp, Wp, Ns, am, bm, acc);
    #pragma unroll 4
    for (; k < K4; k += 4)                      // drain: no prefetch
        wmma_step<false>(Xp, Wp, Ns, am, bm, acc);

    if (K & 3) {                                // only K=131 takes this path
        const int kr = K4 + kofs;
        const float a0 = (kr     < K) ? am : 0.f;
        const float a1 = (kr + 1 < K) ? am : 0.f;
        const float b0 = (kr     < K) ? bm : 0.f;
        const float b1 = (kr + 1 < K) ? bm : 0.f;
        const int k0 = (kr     < K) ? kr     : 0;
        const int k1 = (kr + 1 < K) ? kr + 1 : 0;
        v2f a;
        a.x = X[(size_t)mc * K + k0] * a0;
        a.y = X[(size_t)mc * K + k1] * a1;
        v2f b;
        b.x = W[(size_t)k0 * Ns + nc] * b0;
        b.y = W[(size_t)k1 * Ns + nc] * b1;
        acc = __builtin_amdgcn_wmma_f32_16x16x4_f32(
                  false, a, false, b, (short)0, acc, false, false);
    }

    if (n < N) {
        const float bb    = bias ? bias[n] : 0.f;
        const int   mbase = (lane < 16) ? 0 : 8;
        #pragma unroll
        for (int j = 0; j < 8; ++j) {
            const int mm = mbase + j;
            if (mm < M) {
                float v = acc[j] + bb;
                if (do_relu) v = fmaxf(v, 0.f);
                Y[(size_t)mm * N + n] = v;
            }
        }
    }
}

// ---------------------------------------------------------------------------
// Small glue kernels
// ---------------------------------------------------------------------------
__global__ void copy_f32(float* __restrict__ dst, const float* __restrict__ src, int n)
{
    int i = blockIdx.x * blockDim.x + threadIdx.x;
    if (i < n) dst[i] = src[i];
}

// Face_Rotate_Conv: per-face corner pairs -> conv1d(6->32) -> mean over 3 pairs
__global__ void frc_conv_mean(const float* __restrict__ corner,
                              const float* __restrict__ Wc,
                              const float* __restrict__ bc,
                              float* __restrict__ out)
{
    int t = blockIdx.x * blockDim.x + threadIdx.x;
    if (t >= NFACE * 32) return;
    int f = t >> 5, o = t & 31;
    float s = 0.f;
    #pragma unroll
    for (int p = 0; p < 3; ++p) {
        int q = (p + 1 == 3) ? 0 : p + 1;
        #pragma unroll
        for (int d = 0; d < 3; ++d) {
            s += corner[f * 9 + p * 3 + d] * Wc[o * 6 + d];
            s += corner[f * 9 + q * 3 + d] * Wc[o * 6 + 3 + d];
        }
    }
    out[f * 32 + o] = s * (1.f / 3.f) + bc[o];
}

// kc[f][m] = sum_l exp(-||n_f - K[m][l]||^2) / 32
__global__ void kc_compute(const float* __restrict__ normal,
                           const float* __restrict__ kern,
                           float* __restrict__ kc)
{
    int t = blockIdx.x * blockDim.x + threadIdx.x;
    if (t >= NFACE * 64) return;
    int f = t >> 6, m = t & 63;
    float nx = normal[f * 3 + 0], ny = normal[f * 3 + 1], nz = normal[f * 3 + 2];
    float s = 0.f;
    #pragma unroll
    for (int l = 0; l < 4; ++l) {
        float dx = nx - kern[(m * 4 + l) * 3 + 0];
        float dy = ny - kern[(m * 4 + l) * 3 + 1];
        float dz = nz - kern[(m * 4 + l) * 3 + 2];
        s += expf(-(dx * dx + dy * dy + dz * dz));
    }
    kc[f * 64 + m] = s * (1.f / 32.f);
}

// feat[f] = [ normal_f(3) | kc_f(64) | frc_f(64) ]  -> (8,131)
__global__ void feat_assemble(const float* __restrict__ normal,
                              const float* __restrict__ kc,
                              const float* __restrict__ frc,
                              float* __restrict__ feat)
{
    int t = blockIdx.x * blockDim.x + threadIdx.x;
    if (t >= NFACE * 131) return;
    int f = t / 131, j = t - f * 131;
    float v;
    if (j < 3)       v = normal[f * 3 + j];
    else if (j < 67) v = kc[f * 64 + (j - 3)];
    else             v = frc[f * 64 + (j - 67)];
    feat[t] = v;
}

// agg[f] = (z[f] + z[nb0] + z[nb1] + z[nb2]) / 4
__global__ void neighbor_agg(const float* __restrict__ z,
                             const int* __restrict__ nb,
                             float* __restrict__ out, int C)
{
    int t = blockIdx.x * blockDim.x + threadIdx.x;
    if (t >= NFACE * C) return;
    int f = t / C, j = t - f * C;
    float s = z[f * C + j];
    s += z[nb[f * 3 + 0] * C + j];
    s += z[nb[f * 3 + 1] * C + j];
    s += z[nb[f * 3 + 2] * C + j];
    out[t] = s * 0.25f;
}

// g[j] = max_f h[f*1024 + j]
__global__ void max_pool_faces(const float* __restrict__ h, float* __restrict__ g)
{
    int j = blockIdx.x * blockDim.x + threadIdx.x;
    if (j >= 1024) return;
    float m = h[j];
    #pragma unroll
    for (int f = 1; f < NFACE; ++f) m = fmaxf(m, h[f * 1024 + j]);
    g[j] = m;
}

// ---------------------------------------------------------------------------
extern "C" void kernel_launch(void* const* d_in, const int* in_sizes, int n_in,
                              void* d_out, int out_size, void* d_ws, size_t ws_size,
                              hipStream_t stream)
{
    const float* center  = (const float*)d_in[0];
    const float* corner  = (const float*)d_in[1];
    const float* normal  = (const float*)d_in[2];
    const int*   nb      = (const int*)  d_in[3];
    const float* sp_W1 = (const float*)d_in[4];  const float* sp_b1 = (const float*)d_in[5];
    const float* sp_W2 = (const float*)d_in[6];  const float* sp_b2 = (const float*)d_in[7];
    const float* frc_Wc = (const float*)d_in[8];  const float* frc_bc = (const float*)d_in[9];
    const float* frc_W3 = (const float*)d_in[10]; const float* frc_b3 = (const float*)d_in[11];
    const float* frc_W4 = (const float*)d_in[12]; const float* frc_b4 = (const float*)d_in[13];
    const float* kc_k   = (const float*)d_in[14];
    const float* st_W1 = (const float*)d_in[15]; const float* st_b1 = (const float*)d_in[16];
    const float* st_W2 = (const float*)d_in[17]; const float* st_b2 = (const float*)d_in[18];
    const float* c1_W1 = (const float*)d_in[19]; const float* c1_b1 = (const float*)d_in[20];
    const float* c1_W2 = (const float*)d_in[21]; const float* c1_b2 = (const float*)d_in[22];
    const float* a1_W3 = (const float*)d_in[23]; const float* a1_b3 = (const float*)d_in[24];
    const float* a1_W4 = (const float*)d_in[25]; const float* a1_b4 = (const float*)d_in[26];
    const float* c2_W1 = (const float*)d_in[27]; const float* c2_b1 = (const float*)d_in[28];
    const float* c2_W2 = (const float*)d_in[29]; const float* c2_b2 = (const float*)d_in[30];
    const float* a2_W3 = (const float*)d_in[31]; const float* a2_b3 = (const float*)d_in[32];
    const float* a2_W4 = (const float*)d_in[33]; const float* a2_b4 = (const float*)d_in[34];
    const float* nt_W1 = (const float*)d_in[35]; const float* nt_b1 = (const float*)d_in[36];
    const float* nt_W2 = (const float*)d_in[37]; const float* nt_b2 = (const float*)d_in[38];
    const float* m2_W1 = (const float*)d_in[39]; const float* m2_b1 = (const float*)d_in[40];
    const float* m2_W2 = (const float*)d_in[41]; const float* m2_b2 = (const float*)d_in[42];
    const float* m3_W1 = (const float*)d_in[43]; const float* m3_b1 = (const float*)d_in[44];
    const float* m3_W2 = (const float*)d_in[45]; const float* m3_b2 = (const float*)d_in[46];
    const float* m3_W3 = (const float*)d_in[47]; const float* m3_b3 = (const float*)d_in[48];
    float* out = (float*)d_out;

    // ---- workspace carving (fp32, 256B-aligned blocks) ----
    float* ws = (float*)d_ws;
    size_t off = 0;
    auto alloc = [&](size_t nelem) {
        float* p = ws + off;
        off += (nelem + 63) & ~(size_t)63;
        return p;
    };
    float* t_sp1   = alloc(512);
    float* y_sp    = alloc(512);
    float* meanc   = alloc(256);            // (8,32)
    float* frc_h   = alloc(512);            // (8,64)
    float* frc     = alloc(512);            // (8,64)
    float* kc      = alloc(512);            // (8,64)
    float* feat    = alloc(NFACE * 131);    // (8,131)
    float* z1      = alloc(NFACE * 131);
    float* z       = alloc(NFACE * 131);
    float* in1     = alloc(1560);           // [y_sp | z]
    float* t_c1    = alloc(1600);
    float* out12   = alloc(4096);           // out1 | out2 (adjacent -> in2)
    float* out1    = out12;
    float* out2    = out12 + 2048;
    float* agg1    = alloc(NFACE * 131);
    float* t_a1    = alloc(4096);
    float* t_c2    = alloc(4096);
    float* out34   = alloc(8192);           // out3 | out4 (adjacent -> X5 (8,1024))
    float* out3    = out34;
    float* out4    = out34 + 4096;
    float* agg2    = alloc(NFACE * 256);
    float* t_a2    = alloc(4096);
    float* t_nt    = alloc(8192);           // (8,1024)
    float* out5    = alloc(8192);           // (8,1024)
    float* out6    = alloc(14336);          // [out5 | out3 | out1]
    float* t_m2    = alloc(NFACE * 1024);
    float* h       = alloc(NFACE * 1024);
    float* g       = alloc(1024);
    float* t_m3a   = alloc(512);
    float* t_m3b   = alloc(256);
    if (off * sizeof(float) > ws_size) return;   // not enough scratch

    auto gemm = [&](const float* X, const float* W, const float* b, float* Y,
                    int M, int K, int N, int relu) {
        const int waves_per_block = 8;               // 256 threads = 8 wave32
        int tiles = (N + 15) / 16;
        int grid  = (tiles + waves_per_block - 1) / waves_per_block;
        gemm_wmma_f32<<<grid, waves_per_block * 32, 0, stream>>>(X, W, b, Y, M, K, N, relu);
    };
    auto copy = [&](float* dst, const float* src, int n) {
        copy_f32<<<(n + 255) / 256, 256, 0, stream>>>(dst, src, n);
    };

    // ---- spatial descriptor: 24 -> 512 -> 512 (flattened GEMVs) ----
    gemm(center, sp_W1, sp_b1, t_sp1, 1, 24, 512, 1);
    gemm(t_sp1, sp_W2, sp_b2, y_sp, 1, 512, 512, 0);

    // ---- Face_Rotate_Conv ----
    frc_conv_mean<<<1, 256, 0, stream>>>(corner, frc_Wc, frc_bc, meanc);
    gemm(meanc, frc_W3, frc_b3, frc_h, NFACE, 32, 64, 1);
    gemm(frc_h, frc_W4, frc_b4, frc, NFACE, 64, 64, 0);

    // ---- Kernel correlation + structural descriptor ----
    kc_compute<<<2, 256, 0, stream>>>(normal, kc_k, kc);
    feat_assemble<<<(NFACE * 131 + 255) / 256, 256, 0, stream>>>(normal, kc, frc, feat);
    gemm(feat, st_W1, st_b1, z1, NFACE, 131, 131, 1);
    gemm(z1, st_W2, st_b2, z, NFACE, 131, 131, 0);

    // ---- Mesh1 ----
    copy(in1, y_sp, 512);
    copy(in1 + 512, z, NFACE * 131);
    gemm(in1, c1_W1, c1_b1, t_c1, 1, 1560, 1600, 1);
    gemm(t_c1, c1_W2, c1_b2, out1, 1, 1600, 2048, 0);
    neighbor_agg<<<(NFACE * 131 + 255) / 256, 256, 0, stream>>>(z, nb, agg1, 131);
    gemm(agg1, a1_W3, a1_b3, t_a1, 1, 1048, 4096, 1);
    gemm(t_a1, a1_W4, a1_b4, out2, 1, 4096, 2048, 0);

    // ---- Mesh2 ----
    gemm(out12, c2_W1, c2_b1, t_c2, 1, 4096, 4096, 1);
    gemm(t_c2, c2_W2, c2_b2, out3, 1, 4096, 4096, 0);
    neighbor_agg<<<(NFACE * 256 + 255) / 256, 256, 0, stream>>>(out2, nb, agg2, 256);
    gemm(agg2, a2_W3, a2_b3, t_a2, 1, 2048, 4096, 1);
    gemm(t_a2, a2_W4, a2_b4, out4, 1, 4096, 4096, 0);

    // ---- per-face Net 1024->1024->1024 (M=8 GEMM; X5 = out34 viewed (8,1024)) ----
    gemm(out34, nt_W1, nt_b1, t_nt, NFACE, 1024, 1024, 1);
    gemm(t_nt, nt_W2, nt_b2, out5, NFACE, 1024, 1024, 0);

    // ---- mlp2: 14336 -> 8192 -> 8192 (dominant: 1.05 GB weight stream) ----
    copy(out6, out5, 8192);
    copy(out6 + 8192, out3, 4096);
    copy(out6 + 12288, out1, 2048);
    gemm(out6, m2_W1, m2_b1, t_m2, 1, 14336, 8192, 1);
    gemm(t_m2, m2_W2, m2_b2, h, 1, 8192, 8192, 0);

    // ---- global max-pool over faces + classifier head ----
    max_pool_faces<<<4, 256, 0, stream>>>(h, g);
    gemm(g, m3_W1, m3_b1, t_m3a, 1, 1024, 512, 1);
    gemm(t_m3a, m3_W2, m3_b2, t_m3b, 1, 512, 256, 1);
    gemm(t_m3b, m3_W3, m3_b3, out, 1, 256, 40, 0);
}